// GCN_37847251812925
// MI455X (gfx1250) — compile-verified
//
#include <hip/hip_runtime.h>

// ---------------------------------------------------------------------------
// GCN: h1 = relu(Ahat (x W1) + b1); h2 = relu(Ahat (h1 W2) + b2); out = h2 Wfc + bfc
// Ahat = D^-1/2 (A + I) D^-1/2, per-edge weight dinv[src]*dinv[dst].
// ---------------------------------------------------------------------------

typedef __attribute__((ext_vector_type(2))) float v2f;
typedef __attribute__((ext_vector_type(8))) float v8f;

static __device__ __forceinline__ long long gidx() {
  return (long long)blockIdx.x * blockDim.x + threadIdx.x;
}

// ---- degree / normalization ------------------------------------------------
__global__ void k_deg_init(float* __restrict__ deg, int n) {
  long long i = gidx();
  if (i < n) deg[i] = 1.0f;  // self-loop contributes 1
}

__global__ void k_deg_accum(const int* __restrict__ dst, float* __restrict__ deg, int e) {
  long long i = gidx();
  if (i < e) atomicAdd(&deg[dst[i]], 1.0f);
}

__global__ void k_rsqrt_inplace(float* __restrict__ v, int n) {
  long long i = gidx();
  if (i < n) v[i] = rsqrtf(v[i]);
}

// ---- WMMA f32 GEMM: Out[M,Cout] = X[M,K] @ W[K,Cout] (+bias) ---------------
// W is staged in LDS once per block, PRE-SWIZZLED so each lane's B fragment
// (two consecutive K values for one column) is a contiguous, 8B-aligned pair:
//   sW[(((k/4)*2 + (k%4)/2)*Cout + n)*2 + (k%2)] = W[k][n]
// -> in-loop B fetch is a single ds_load_b64 at k0*Cout + half*2*Cout +
//    (nt*16+l15)*2, with nt folding into an immediate DS offset (nt*128 B).
// Each wave owns one 16-row M-tile and sweeps all NT 16-col N-tiles per
// k-step, reusing one A fragment for NT independent WMMA accumulator chains.
// Optional fused epilogue: Agg[m,n] = Out[m,n] * dinv[m]^2 (self-loop term).
template <int NT>
__global__ void k_gemm_wmma_f32(const float* __restrict__ X,
                                const float* __restrict__ W,
                                const float* __restrict__ bias,  // may be null
                                const float* __restrict__ dinv,  // used if Agg
                                float* __restrict__ Out,
                                float* __restrict__ Agg,         // may be null
                                int M, int K) {
  constexpr int Cout = NT * 16;
  __shared__ float sW[96 * Cout];  // K <= 96

  // Cooperative swizzled stage of W[K, Cout] into LDS. All waves participate;
  // barrier BEFORE any wave can skip on the M-tile bound.
  for (int i = threadIdx.x; i < K * Cout; i += blockDim.x) {
    const int k = i / Cout;
    const int n = i - k * Cout;
    const int dstIdx = (((k >> 2) * 2 + ((k & 3) >> 1)) * Cout + n) * 2 + (k & 1);
    sW[dstIdx] = W[i];
  }
  __syncthreads();

  const int lane = threadIdx.x & 31;
  const int wave = threadIdx.x >> 5;
  const int mtile = blockIdx.x * (blockDim.x >> 5) + wave;
  if (mtile < (M >> 4)) {  // wave-uniform: EXEC all-1s inside
    const int half = lane >> 4;   // 0: lanes 0-15, 1: lanes 16-31
    const int l15  = lane & 15;
    const int kb   = half * 2;    // K sub-offset per ISA 16x4 f32 layout
    const int m0   = mtile << 4;

    v8f acc[NT] = {};
    const float* arow  = X + (size_t)(m0 + l15) * K + kb;        // A frag (b64)
    const float* bbase = sW + half * (2 * Cout) + l15 * 2;       // B frag base
    for (int k0 = 0; k0 < K; k0 += 4) {
      v2f a;
      a.x = arow[k0];
      a.y = arow[k0 + 1];
      const float* brow = bbase + k0 * Cout;
#pragma unroll
      for (int nt = 0; nt < NT; ++nt) {
        const v2f b = *(const v2f*)(brow + nt * 32);  // ds_load_b64, imm offset
        acc[nt] = __builtin_amdgcn_wmma_f32_16x16x4_f32(
            /*neg_a=*/false, a, /*neg_b=*/false, b,
            /*c_mod=*/(short)0, acc[nt], /*reuse_a=*/false, /*reuse_b=*/false);
      }
    }

    // C/D layout: VGPR v -> row m0 + half*8 + v, column nt*16 + l15
    const int mb = m0 + half * 8;
    float dv2[8];
    if (Agg) {
#pragma unroll
      for (int v = 0; v < 8; ++v) {
        const float d = dinv[mb + v];
        dv2[v] = d * d;
      }
    }
#pragma unroll
    for (int nt = 0; nt < NT; ++nt) {
      const int n = nt * 16 + l15;
      const float bv = bias ? bias[n] : 0.0f;
      float* orow = Out + (size_t)mb * Cout + n;
#pragma unroll
      for (int v = 0; v < 8; ++v) {
        const float val = acc[nt][v] + bv;
        orow[(size_t)v * Cout] = val;
        if (Agg) Agg[(size_t)(mb + v) * Cout + n] = val * dv2[v];
      }
    }
  }
}

// ---- aggregation -----------------------------------------------------------
// One thread per (edge, 4-channel group): float4 gather + 4 f32 atomics.
// Feature matrix (19.2 MB) is L2-resident (192 MB), so these hit L2.
__global__ void k_scatter_edges(const float* __restrict__ T,   // [N,C]
                                const int* __restrict__ src,
                                const int* __restrict__ dst,
                                const float* __restrict__ dinv,
                                float* __restrict__ Agg,        // [N,C]
                                int E, int C) {
  const int groups = C >> 2;
  long long idx = gidx();
  long long total = (long long)E * groups;
  if (idx >= total) return;
  const int e = (int)(idx / groups);
  const int g = (int)(idx - (long long)e * groups);
  const int s = src[e];
  const int d = dst[e];
  const float w = dinv[s] * dinv[d];
  const float4 m = *(const float4*)(T + (size_t)s * C + g * 4);
  float* out = Agg + (size_t)d * C + g * 4;
  atomicAdd(out + 0, m.x * w);
  atomicAdd(out + 1, m.y * w);
  atomicAdd(out + 2, m.z * w);
  atomicAdd(out + 3, m.w * w);
}

__global__ void k_bias_relu(float* __restrict__ H, const float* __restrict__ b,
                            long long total, int C) {
  long long idx = gidx();
  if (idx >= total) return;
  const int c = (int)(idx % C);
  H[idx] = fmaxf(H[idx] + b[c], 0.0f);
}

// ---------------------------------------------------------------------------
extern "C" void kernel_launch(void* const* d_in, const int* in_sizes, int n_in,
                              void* d_out, int out_size, void* d_ws, size_t ws_size,
                              hipStream_t stream) {
  const float* x    = (const float*)d_in[0];
  const int*   ei   = (const int*)d_in[1];
  const float* W1   = (const float*)d_in[2];
  const float* b1   = (const float*)d_in[3];
  const float* W2   = (const float*)d_in[4];
  const float* b2   = (const float*)d_in[5];
  const float* Wfc  = (const float*)d_in[6];
  const float* bfc  = (const float*)d_in[7];
  float* out = (float*)d_out;

  const int IN_C = 64, HID = 96, OUT_C = 32;
  const int N = in_sizes[0] / IN_C;
  const int E = in_sizes[1] / 2;
  const int* src = ei;
  const int* dst = ei + E;

  // workspace: dinv[N] | bufA[N*HID] (t) | bufB[N*HID] (h1) | bufC[N*HID]
  float* dinv = (float*)d_ws;
  float* bufA = dinv + N;
  float* bufB = bufA + (size_t)N * HID;
  float* bufC = bufB + (size_t)N * HID;

  const int B = 256;
  const long long NH = (long long)N * HID;
  const long long EG = (long long)E * (HID >> 2);

  dim3 blk(B);
  auto blocks = [](long long n, int per) { return dim3((unsigned)((n + per - 1) / per)); };

  const int mtiles = N >> 4;                // 3125
  const dim3 gemmGrid = blocks(mtiles, 8);  // 8 waves (M-tiles) per 256-thr block

  // --- normalization ---
  k_deg_init<<<blocks(N, B), blk, 0, stream>>>(dinv, N);
  k_deg_accum<<<blocks(E, B), blk, 0, stream>>>(dst, dinv, E);
  k_rsqrt_inplace<<<blocks(N, B), blk, 0, stream>>>(dinv, N);

  // --- layer 1: t1 = x@W1 (bufA) with fused self-loop init (bufB) ---
  k_gemm_wmma_f32<6><<<gemmGrid, blk, 0, stream>>>(x, W1, nullptr, dinv, bufA, bufB, N, IN_C);
  k_scatter_edges<<<blocks(EG, B), blk, 0, stream>>>(bufA, src, dst, dinv, bufB, E, HID);
  k_bias_relu<<<blocks(NH, B), blk, 0, stream>>>(bufB, b1, NH, HID);

  // --- layer 2: t2 = h1@W2 (bufA) with fused self-loop init (bufC) ---
  k_gemm_wmma_f32<6><<<gemmGrid, blk, 0, stream>>>(bufB, W2, nullptr, dinv, bufA, bufC, N, HID);
  k_scatter_edges<<<blocks(EG, B), blk, 0, stream>>>(bufA, src, dst, dinv, bufC, E, HID);
  k_bias_relu<<<blocks(NH, B), blk, 0, stream>>>(bufC, b2, NH, HID);

  // --- FC head: out = h2 @ Wfc + bfc ---
  k_gemm_wmma_f32<2><<<gemmGrid, blk, 0, stream>>>(bufC, Wfc, bfc, nullptr, out, nullptr, N, HID);
}